// triplet_tagger_90795608637572
// MI455X (gfx1250) — compile-verified
//
#include <hip/hip_runtime.h>
#include <hip/hip_bf16.h>

typedef float v2f __attribute__((ext_vector_type(2)));
typedef float v8f __attribute__((ext_vector_type(8)));

namespace {

constexpr int kWaves        = 4;   // wave32 waves per block
constexpr int kTilesPerWave = 2;   // 16-event tiles handled sequentially per wave
constexpr int kThreads      = kWaves * 32;

// ---------------- LDS layout (floats) ----------------
// Weights are stored K-pair-interleaved:  wp[(k>>1)*(2*NPAD) + 2*n + (k&1)]
// so a B fragment (rows k,k+1 at column n) is one aligned 8-byte load.
constexpr int WLC_OFF = 0;                   // 12 Wl col-1 coeffs + bl[1] -> 16
constexpr int W1P_OFF = 16;                  // W1^T paired: 10 pair-rows x 160 = 1600
constexpr int B1_OFF  = W1P_OFF + 20 * 80;   // 80
constexpr int W2P_OFF = B1_OFF + 80;         // W2^T paired: 40 pair-rows x 96 = 3840 (cols 40..47 = 0)
constexpr int B2_OFF  = W2P_OFF + 80 * 48;   // 48 (padded)
constexpr int W3P_OFF = B2_OFF + 48;         // W3^T paired: 20 pair-rows x 96 = 1920 (cols 21..47 = 0)
constexpr int B3_OFF  = W3P_OFF + 40 * 48;   // 48 (padded)
constexpr int SCR_OFF = B3_OFF + 48;

// per-wave scratch: region1 holds s (16x20) then h2 (16x52-padded); region2 = h1 (16x84-padded)
constexpr int S_STRIDE  = 20;
constexpr int H1_STRIDE = 84;   // 80 + pad: conflict-free A-frag reads & D stores
constexpr int H2_STRIDE = 52;   // 48 + pad
constexpr int R1_SIZE   = 16 * H2_STRIDE;             // 832
constexpr int WAVE_SCR  = R1_SIZE + 16 * H1_STRIDE;   // 832 + 1344 = 2176
constexpr int LDS_FLOATS = SCR_OFF + kWaves * WAVE_SCR;
static_assert(LDS_FLOATS * 4 <= 65536, "static LDS budget exceeded");
static_assert((SCR_OFF % 2) == 0 && (WAVE_SCR % 2) == 0 && (R1_SIZE % 2) == 0,
              "8B alignment of scratch regions");

// ---- WMMA 16x16x4 f32 fragment loaders (ISA 7.12.2 layouts) ----
// A (16x4): lanes 0-15 -> M=lane, VGPRs {K0,K1}; lanes 16-31 -> {K2,K3}.
// K pairs are contiguous in the activation buffers -> single ds_load_b64.
__device__ __forceinline__ v2f lds_afrag(const float* __restrict__ buf, int stride, int k0,
                                         int lane) {
  const int m    = lane & 15;
  const int koff = (lane >> 4) << 1;
  return *reinterpret_cast<const v2f*>(buf + m * stride + k0 + koff);
}

// B (4x16): lanes 0-15 -> N=lane, VGPRs {row K0, row K1}; lanes 16-31 -> {K2,K3}.
// Pair-interleaved weight layout -> single ds_load_b64 per fragment.
__device__ __forceinline__ v2f lds_bfrag(const float* __restrict__ wp, int npad2, int k0, int c0,
                                         int lane) {
  const int n  = lane & 15;
  const int kp = (k0 >> 1) + (lane >> 4);
  return *reinterpret_cast<const v2f*>(wp + kp * npad2 + ((c0 + n) << 1));
}

// D = A(16xK) * W^T(KxN-tiles) + bias, ReLU, store to LDS row-major.
// C/D layout: lane&15 = column-in-tile; VGPR r = row r (lanes 0-15) / r+8 (lanes 16-31).
template <int KTOT, int NTILES>
__device__ __forceinline__ void fc_layer_relu(const float* __restrict__ abuf, int astride,
                                              const float* __restrict__ wp, int npad2,
                                              const float* __restrict__ bias,
                                              float* __restrict__ hbuf, int hstride, int lane) {
  const int n      = lane & 15;
  const int rowoff = (lane >> 4) << 3;
#pragma unroll
  for (int t = 0; t < NTILES; ++t) {
    const int c0   = t * 16;
    const float bv = bias[c0 + n];
    v8f acc;
#pragma unroll
    for (int r = 0; r < 8; ++r) acc[r] = bv;   // bias folded into accumulator init
#pragma unroll
    for (int k0 = 0; k0 < KTOT; k0 += 4) {
      v2f a = lds_afrag(abuf, astride, k0, lane);
      v2f b = lds_bfrag(wp, npad2, k0, c0, lane);
      acc = __builtin_amdgcn_wmma_f32_16x16x4_f32(false, a, false, b, (short)0, acc,
                                                  false, false);
    }
#pragma unroll
    for (int r = 0; r < 8; ++r) {
      float v = acc[r];
      hbuf[(rowoff + r) * hstride + c0 + n] = v > 0.f ? v : 0.f;
    }
  }
}

} // namespace

__global__ __launch_bounds__(kThreads) void triplet_tagger_kernel(
    const float* __restrict__ x,  const float* __restrict__ Wl, const float* __restrict__ bl,
    const float* __restrict__ W1, const float* __restrict__ b1,
    const float* __restrict__ W2, const float* __restrict__ b2,
    const float* __restrict__ W3, const float* __restrict__ b3,
    float* __restrict__ out, int n_events) {
  __shared__ __align__(16) float smem[LDS_FLOATS];
  const int tid = threadIdx.x;

  // ---- cooperative weight staging into K-pair-interleaved layout ----
  if (tid < 12)  smem[WLC_OFF + tid] = Wl[tid * 2 + 1];   // column 1 of Wl
  if (tid == 12) smem[WLC_OFF + 12]  = bl[1];
  for (int i = tid; i < 20 * 80; i += kThreads) {          // W1^T paired [10][80][2]
    const int k = i / 80, n = i % 80;
    smem[W1P_OFF + (k >> 1) * 160 + (n << 1) + (k & 1)] = W1[n * 20 + k];
  }
  for (int i = tid; i < 80; i += kThreads) smem[B1_OFF + i] = b1[i];
  for (int i = tid; i < 80 * 48; i += kThreads) {          // W2^T paired [40][48][2], zero-pad cols
    const int k = i / 48, n = i % 48;
    smem[W2P_OFF + (k >> 1) * 96 + (n << 1) + (k & 1)] = (n < 40) ? W2[n * 80 + k] : 0.f;
  }
  for (int i = tid; i < 48; i += kThreads) smem[B2_OFF + i] = (i < 40) ? b2[i] : 0.f;
  for (int i = tid; i < 40 * 48; i += kThreads) {          // W3^T paired [20][48][2], zero-pad cols
    const int k = i / 48, n = i % 48;
    smem[W3P_OFF + (k >> 1) * 96 + (n << 1) + (k & 1)] = (n < 21) ? W3[n * 40 + k] : 0.f;
  }
  for (int i = tid; i < 48; i += kThreads) smem[B3_OFF + i] = (i < 21) ? b3[i] : 0.f;
  __syncthreads();

  const int lane = tid & 31;
  const int wave = tid >> 5;
  float* sw = smem + SCR_OFF + wave * WAVE_SCR;  // region1: s, later h2
  float* h1 = sw + R1_SIZE;                      // region2: h1

  float wlr[12];
#pragma unroll
  for (int f = 0; f < 12; ++f) wlr[f] = smem[WLC_OFF + f];
  const float bl1 = smem[WLC_OFF + 12];

  const int tile0 = blockIdx.x * (kWaves * kTilesPerWave) + wave * kTilesPerWave;
  for (int it = 0; it < kTilesPerWave; ++it) {
    const int ev_base = (tile0 + it) * 16;

    // ---- phase 1: Lorentz head scores s[16][20] = x·Wl[:,1] + bl[1] ----
    // 320 (event,triplet) dots over 32 lanes; 3x float4 loads each (streams x at HBM rate)
#pragma unroll
    for (int i = 0; i < 10; ++i) {
      const int p   = lane + 32 * i;
      const int evl = p / 20;
      const int t   = p - evl * 20;
      const int ev  = ev_base + evl;
      float acc = 0.f;
      if (ev < n_events) {
        const float4* xp = reinterpret_cast<const float4*>(x + (size_t)ev * 240 + t * 12);
        const float4 q0 = xp[0], q1 = xp[1], q2 = xp[2];
        acc  = bl1;
        acc += q0.x * wlr[0] + q0.y * wlr[1] + q0.z * wlr[2]  + q0.w * wlr[3];
        acc += q1.x * wlr[4] + q1.y * wlr[5] + q1.z * wlr[6]  + q1.w * wlr[7];
        acc += q2.x * wlr[8] + q2.y * wlr[9] + q2.z * wlr[10] + q2.w * wlr[11];
      }
      sw[evl * S_STRIDE + t] = acc;
    }
    asm volatile("s_wait_dscnt 0" ::: "memory");  // cross-lane LDS RAW within wave

    // ---- layer 1: [16x20] x [20x80] -> h1, ReLU (5 N-tiles x 5 K-steps) ----
    fc_layer_relu<20, 5>(sw, S_STRIDE, smem + W1P_OFF, 160, smem + B1_OFF, h1, H1_STRIDE, lane);
    asm volatile("s_wait_dscnt 0" ::: "memory");

    // ---- layer 2: [16x80] x [80x48pad] -> h2 (reuses region1), ReLU ----
    fc_layer_relu<80, 3>(h1, H1_STRIDE, smem + W2P_OFF, 96, smem + B2_OFF, sw, H2_STRIDE, lane);
    asm volatile("s_wait_dscnt 0" ::: "memory");

    // ---- layer 3: [16x40] x [40x32pad] -> out[ev][0..20] (no ReLU) ----
    {
      const int n      = lane & 15;
      const int rowoff = (lane >> 4) << 3;
#pragma unroll
      for (int t3 = 0; t3 < 2; ++t3) {
        const int c0   = t3 * 16;
        const float bv = smem[B3_OFF + c0 + n];
        v8f acc;
#pragma unroll
        for (int r = 0; r < 8; ++r) acc[r] = bv;
#pragma unroll
        for (int k0 = 0; k0 < 40; k0 += 4) {
          v2f a = lds_afrag(sw, H2_STRIDE, k0, lane);
          v2f b = lds_bfrag(smem + W3P_OFF, 96, k0, c0, lane);
          acc = __builtin_amdgcn_wmma_f32_16x16x4_f32(false, a, false, b, (short)0, acc,
                                                      false, false);
        }
        const int col = c0 + n;
#pragma unroll
        for (int r = 0; r < 8; ++r) {   // lanes 0-15 write 16 consecutive floats of one row
          const int ev = ev_base + rowoff + r;
          if (col < 21 && ev < n_events) out[(size_t)ev * 21 + col] = acc[r];
        }
      }
    }
  }
}

extern "C" void kernel_launch(void* const* d_in, const int* in_sizes, int n_in,
                              void* d_out, int out_size, void* d_ws, size_t ws_size,
                              hipStream_t stream) {
  (void)n_in; (void)out_size; (void)d_ws; (void)ws_size;
  const float* x  = (const float*)d_in[0];
  const float* Wl = (const float*)d_in[1];
  const float* bl = (const float*)d_in[2];
  const float* W1 = (const float*)d_in[3];
  const float* b1 = (const float*)d_in[4];
  const float* W2 = (const float*)d_in[5];
  const float* b2 = (const float*)d_in[6];
  const float* W3 = (const float*)d_in[7];
  const float* b3 = (const float*)d_in[8];
  float* out = (float*)d_out;

  const int n_events = in_sizes[0] / 240;                    // [N,20,4,3]
  const int tiles    = (n_events + 15) / 16;
  const int tiles_per_block = kWaves * kTilesPerWave;
  const int grid = (tiles + tiles_per_block - 1) / tiles_per_block;

  triplet_tagger_kernel<<<grid, kThreads, 0, stream>>>(x, Wl, bl, W1, b1, W2, b2, W3, b3,
                                                       out, n_events);
}